// BSplineCurve_38259568673546
// MI455X (gfx1250) — compile-verified
//
#include <hip/hip_runtime.h>

// B-spline curve eval, densified per-m into a (64b x 64c) * (64c x 32d) f32 GEMM
// so the contraction runs on V_WMMA_F32_16X16X4_F32 (CDNA5 / gfx1250, wave32).
// Memory-bound: ~296 MB HBM traffic -> ~13 us floor @ 23.3 TB/s.

typedef float v2f __attribute__((ext_vector_type(2)));
typedef float v8f __attribute__((ext_vector_type(8)));

#define BDIM  512
#define MDIM  4096
#define CDIM  64
#define DDIM  32
#define DEG   3
#define NKNOT (CDIM + DEG + 1)   // 68
#define BT    64                 // batch rows per block
#define AST   68                 // padded LDS row stride for W (floats); 68%64=4 kills bank conflicts

__global__ __launch_bounds__(256)
void bspline_wmma_kernel(const float* __restrict__ u,
                         const float* __restrict__ cp,
                         const float* __restrict__ knots,
                         float* __restrict__ out)
{
    __shared__ __attribute__((aligned(16))) float cp_lds[CDIM * DDIM]; // [c][d], 8 KB
    __shared__ float W_lds[BT * AST];                                  // [b][c] padded, ~17 KB
    __shared__ float kn[NKNOT];

    const int tid = threadIdx.x;
    const int m   = blockIdx.y;
    const int b0  = blockIdx.x * BT;

    // ---- Phase 0: stage control_points[m] (64x32 f32) and knots into LDS ----
    const float4* gcp4 = (const float4*)(cp + (size_t)m * (CDIM * DDIM));
    float4* cl4 = (float4*)cp_lds;
    cl4[tid]       = gcp4[tid];          // 512 float4 total, 256 threads x2
    cl4[tid + 256] = gcp4[tid + 256];
    if (tid < NKNOT) kn[tid] = knots[tid];

    // ---- Phase 1: zero the dense weight matrix W ----
    for (int i = tid; i < BT * AST; i += 256) W_lds[i] = 0.0f;

    __syncthreads();

    // ---- Phase 2: 64 threads -> span + cubic Cox-de Boor basis, scatter into W ----
    if (tid < BT) {
        const int b  = b0 + tid;
        const float uv = u[(size_t)b * MDIM + m];
        float un = uv / (1.0f + fabsf(uv));
        const float kmin = kn[DEG];       // == -1
        const float kmax = kn[CDIM];      // == +1
        un = kmin + (un + 1.0f) * 0.5f * (kmax - kmin);

        // i = clip(searchsorted(knots, un, 'right') - 1, DEG, CDIM-1)
        int i = DEG;
        for (int j = DEG + 1; j <= CDIM - 1; ++j)
            if (kn[j] <= un) i = j;

        // left[q] = un - knots[i+1-q], right[q] = knots[i+q] - un  (q = 1..DEG)
        float left[DEG + 1], right[DEG + 1];
        #pragma unroll
        for (int j = 1; j <= DEG; ++j) {
            left[j]  = un - kn[i + 1 - j];
            right[j] = kn[i + j] - un;
        }
        float N[DEG + 1];
        N[0] = 1.0f;
        #pragma unroll
        for (int j = 1; j <= DEG; ++j) {
            float saved = 0.0f;
            #pragma unroll
            for (int r = 0; r < j; ++r) {
                const float temp = N[r] / (right[r + 1] + left[j - r]);
                N[r] = saved + right[r + 1] * temp;
                saved = left[j - r] * temp;
            }
            N[j] = saved;
        }
        float* Wrow = &W_lds[tid * AST];
        Wrow[i - 3] = N[0];
        Wrow[i - 2] = N[1];
        Wrow[i - 1] = N[2];
        Wrow[i    ] = N[3];
    }

    __syncthreads();

    // ---- Phase 3: 8 waves, one 16x16 output tile each; K=64 as 16 chained WMMAs ----
    const int lane = tid & 31;
    const int wave = tid >> 5;
    const int rt   = wave >> 1;      // row tile 0..3 (b)
    const int nt   = wave & 1;       // col tile 0..1 (d)
    const int half = lane >> 4;      // lane half: 0 or 1
    const int l16  = lane & 15;

    v8f acc = {};
    const int arow = (rt * 16 + l16) * AST;
    #pragma unroll
    for (int k = 0; k < 16; ++k) {
        // A (16x4 f32): v0 = {K0 | K2}, v1 = {K1 | K3} across lane halves
        const int kb = k * 4 + half * 2;
        v2f a, bf;
        a.x  = W_lds[arow + kb];
        a.y  = W_lds[arow + kb + 1];
        // B (4x16 f32), mirrored layout: v0 = {K0 row | K2 row}, v1 = {K1 | K3}
        bf.x = cp_lds[(kb)     * DDIM + nt * 16 + l16];
        bf.y = cp_lds[(kb + 1) * DDIM + nt * 16 + l16];
        acc = __builtin_amdgcn_wmma_f32_16x16x4_f32(
                  /*neg_a=*/false, a, /*neg_b=*/false, bf,
                  /*c_mod=*/(short)0, acc, /*reuse_a=*/false, /*reuse_b=*/false);
    }

    // ---- Phase 4: store 16x16 f32 C/D tile: VGPR r -> rows r (lanes 0-15), r+8 (16-31)
    float* op = out + ((size_t)(b0 + rt * 16 + half * 8) * MDIM + m) * DDIM + nt * 16 + l16;
    #pragma unroll
    for (int r = 0; r < 8; ++r)
        op[(size_t)r * (MDIM * DDIM)] = acc[r];
}

extern "C" void kernel_launch(void* const* d_in, const int* in_sizes, int n_in,
                              void* d_out, int out_size, void* d_ws, size_t ws_size,
                              hipStream_t stream) {
    (void)in_sizes; (void)n_in; (void)out_size; (void)d_ws; (void)ws_size;
    const float* u     = (const float*)d_in[0];   // (B, M)
    const float* cp    = (const float*)d_in[1];   // (M, C, D)
    const float* knots = (const float*)d_in[2];   // (C+DEG+1,)
    float* out = (float*)d_out;                   // (B, M, D)

    dim3 grid(BDIM / BT, MDIM);   // (8, 4096)
    dim3 block(256);
    bspline_wmma_kernel<<<grid, block, 0, stream>>>(u, cp, knots, out);
}